// GNNEncoder_9826885173840
// MI455X (gfx1250) — compile-verified
//
#include <hip/hip_runtime.h>
#include <hip/hip_bf16.h>

// ---------------------------------------------------------------------------
// GNN encoder for MI455X (gfx1250):
//  - fp32 WMMA (v_wmma_f32_16x16x4_f32) for the SAGE GEMMs (fused LayerNorm)
//  - CSR-bucketed, atomic-free softmax aggregation (per-node gather, b128,
//    global_prefetch for the data-dependent row chase)
//  - skip path algebraically reduced to a 256-vector matvec
// ---------------------------------------------------------------------------

#define NN     50000
#define EE     1600000
#define DD     256
#define KK     512          // 2*D for the SAGE concat GEMM
#define LL     3
#define EPS_LN  1e-5f
#define EPS_DEN 1e-8f

#define SCAN_CHUNK 256
#define NCHUNKS    ((NN + SCAN_CHUNK - 1) / SCAN_CHUNK)   // 196

typedef float v2f __attribute__((ext_vector_type(2)));
typedef float v8f __attribute__((ext_vector_type(8)));

__device__ __forceinline__ void atomic_add_f32(float* p, float v) {
    unsafeAtomicAdd(p, v);   // native global_atomic_add_f32 on gfx1250
}

// ===========================================================================
// CSR build (once per launch): bucket edges by dst.
// ===========================================================================

__global__ __launch_bounds__(256)
void count_deg_kernel(const int* __restrict__ dst, int* __restrict__ deg, int e) {
    int i = blockIdx.x * blockDim.x + threadIdx.x;
    if (i >= e) return;
    atomicAdd(&deg[dst[i]], 1);
}

// Per-chunk exclusive scan (Hillis-Steele in LDS); emits chunk totals.
__global__ __launch_bounds__(256)
void scan1_kernel(const int* __restrict__ deg, int* __restrict__ rowptr,
                  int* __restrict__ chunk_sum, int n) {
    __shared__ int s[SCAN_CHUNK];
    int t = threadIdx.x;
    int i = blockIdx.x * SCAN_CHUNK + t;
    int v = (i < n) ? deg[i] : 0;
    s[t] = v;
    __syncthreads();
#pragma unroll
    for (int off = 1; off < SCAN_CHUNK; off <<= 1) {
        int x = (t >= off) ? s[t - off] : 0;
        __syncthreads();
        s[t] += x;
        __syncthreads();
    }
    if (i < n) rowptr[i] = s[t] - v;              // exclusive within chunk
    if (t == SCAN_CHUNK - 1) chunk_sum[blockIdx.x] = s[t];
}

// Single-block exclusive scan of chunk totals (in place).
__global__ __launch_bounds__(256)
void scan2_kernel(int* __restrict__ chunk_sum, int nchunks) {
    __shared__ int s[SCAN_CHUNK];
    int t = threadIdx.x;
    int v = (t < nchunks) ? chunk_sum[t] : 0;
    s[t] = v;
    __syncthreads();
#pragma unroll
    for (int off = 1; off < SCAN_CHUNK; off <<= 1) {
        int x = (t >= off) ? s[t - off] : 0;
        __syncthreads();
        s[t] += x;
        __syncthreads();
    }
    if (t < nchunks) chunk_sum[t] = s[t] - v;     // exclusive chunk offsets
}

__global__ __launch_bounds__(256)
void scan3_kernel(int* __restrict__ rowptr, const int* __restrict__ chunk_sum,
                  int n, int e) {
    int i = blockIdx.x * blockDim.x + threadIdx.x;
    if (i < n) rowptr[i] += chunk_sum[i >> 8];
    if (i == 0) rowptr[n] = e;
}

__global__ __launch_bounds__(256)
void fill_csr_kernel(const int* __restrict__ dst, int* __restrict__ pos,
                     int* __restrict__ eid, int e) {
    int i = blockIdx.x * blockDim.x + threadIdx.x;
    if (i >= e) return;
    int p = atomicAdd(&pos[dst[i]], 1);
    eid[p] = i;
}

// ===========================================================================
// Per-layer kernels
// ===========================================================================

// s_dst[n] = h[n]·w[:256], s_src[n] = h[n]·w[256:]  (one wave32 per node)
__global__ __launch_bounds__(256)
void attn_scores_kernel(const float* __restrict__ h,
                        const float* __restrict__ w_attn,   // 512 floats
                        float* __restrict__ s_dst,
                        float* __restrict__ s_src,
                        int n) {
    int wave = (blockIdx.x * blockDim.x + threadIdx.x) >> 5;
    int lane = threadIdx.x & 31;
    if (wave >= n) return;
    const float* row = h + (size_t)wave * DD;
    float a1 = 0.f, a2 = 0.f;
#pragma unroll
    for (int j = 0; j < 8; ++j) {
        int c = lane + 32 * j;
        float v = row[c];
        a1 += v * w_attn[c];
        a2 += v * w_attn[DD + c];
    }
#pragma unroll
    for (int off = 16; off > 0; off >>= 1) {
        a1 += __shfl_xor(a1, off, 32);
        a2 += __shfl_xor(a2, off, 32);
    }
    if (lane == 0) { s_dst[wave] = a1; s_src[wave] = a2; }
}

// a[e] = exp(s_dst[dst]+s_src[src]+b)  (no atomics)
__global__ __launch_bounds__(256)
void edge_exp_kernel(const int* __restrict__ src,
                     const int* __restrict__ dst,
                     const float* __restrict__ s_dst,
                     const float* __restrict__ s_src,
                     const float* __restrict__ b_attn, int layer,
                     float* __restrict__ a_edge,
                     int e) {
    int i = blockIdx.x * blockDim.x + threadIdx.x;
    if (i >= e) return;
    a_edge[i] = __expf(s_dst[dst[i]] + s_src[src[i]] + b_attn[layer]);
}

// Atomic-free aggregation: one wave32 per node.
//   denom = sum_{e in in(n)} a[e]  (shuffle-tree reduce)
//   agg[n,:] = sum_e (a[e]/max(denom,eps)) * h[src_e,:]   (float4 gathers)
// The next edge's source row is software-prefetched (global_prefetch_b8) to
// hide the data-dependent L2 pointer chase behind the current edge's FMAs.
__global__ __launch_bounds__(256)
void node_agg_kernel(const int* __restrict__ rowptr,
                     const int* __restrict__ eid,
                     const int* __restrict__ src,
                     const float* __restrict__ a_edge,
                     const float* __restrict__ h,
                     float* __restrict__ agg,
                     int n) {
    int node = (blockIdx.x * blockDim.x + threadIdx.x) >> 5;
    int lane = threadIdx.x & 31;
    if (node >= n) return;
    int jb = rowptr[node], je = rowptr[node + 1];

    // denominator (lanes parallel over edges)
    float d = 0.f;
    for (int j = jb + lane; j < je; j += 32) d += a_edge[eid[j]];
#pragma unroll
    for (int off = 16; off > 0; off >>= 1) d += __shfl_xor(d, off, 32);
    float inv = 1.0f / fmaxf(d, EPS_DEN);

    // gather-accumulate (lanes parallel over features: 2 float4 each)
    float ax = 0.f, ay = 0.f, az = 0.f, aw = 0.f;
    float bx = 0.f, by = 0.f, bz = 0.f, bw = 0.f;
    for (int j = jb; j < je; ++j) {
        int e = eid[j];                       // uniform across the wave
        float alpha = a_edge[e] * inv;
        const float4* hr = (const float4*)(h + (size_t)src[e] * DD);
        if (j + 1 < je) {                     // prefetch next source row
            const float* hn = h + (size_t)src[eid[j + 1]] * DD;
            __builtin_prefetch(hn + 4 * lane, 0, 3);
            __builtin_prefetch(hn + 128 + 4 * lane, 0, 3);
        }
        float4 v0 = hr[lane];
        float4 v1 = hr[lane + 32];
        ax += alpha * v0.x; ay += alpha * v0.y; az += alpha * v0.z; aw += alpha * v0.w;
        bx += alpha * v1.x; by += alpha * v1.y; bz += alpha * v1.z; bw += alpha * v1.w;
    }
    float4* ar = (float4*)(agg + (size_t)node * DD);
    float4 o0; o0.x = ax; o0.y = ay; o0.z = az; o0.w = aw;
    float4 o1; o1.x = bx; o1.y = by; o1.z = bz; o1.w = bw;
    ar[lane]      = o0;
    ar[lane + 32] = o1;
}

// ===========================================================================
// Fused SAGE GEMM + LayerNorm + ReLU (fp32 WMMA).
//   z = [h | agg] @ W (512x256) + bias ;  h_out = relu(gamma*norm(z)+beta)
// Block: 256 threads (8 waves), 16 rows x 256 cols; A-tile staged in padded
// LDS with b128 loads/stores; each wave owns two 16x16 col tiles ->
// 256 v_wmma_f32_16x16x4_f32 per wave.
// ===========================================================================
__global__ __launch_bounds__(256)
void sage_gemm_ln_kernel(const float* __restrict__ h,
                         const float* __restrict__ agg,
                         const float* __restrict__ W,      // 512 x 256 row-major
                         const float* __restrict__ bias,   // 256
                         const float* __restrict__ gamma,  // 256
                         const float* __restrict__ beta,   // 256
                         float* __restrict__ hout,
                         int n) {
    __shared__ float sA[16][KK + 4];   // row stride 516 floats (16B aligned)
    __shared__ float zs[16][DD + 4];
    __shared__ float red1[16][16];
    __shared__ float red2[16][16];

    const int row0 = blockIdx.x * 16;
    const int t = threadIdx.x;

    // Stage [h | agg] tile as float4 (8 x b128 per thread).
    for (int i = t; i < 16 * (KK / 4); i += 256) {
        int r  = i >> 7;              // / 128
        int c4 = i & 127;             // float4 column
        int c  = c4 * 4;
        float4 v = (c < DD)
            ? *(const float4*)(h   + (size_t)(row0 + r) * DD + c)
            : *(const float4*)(agg + (size_t)(row0 + r) * DD + (c - DD));
        *(float4*)&sA[r][c] = v;
    }
    __syncthreads();

    const int wid  = t >> 5;
    const int lane = t & 31;
    const int lrow = lane & 15;
    const int koff = (lane < 16) ? 0 : 2;
    const int col0 = wid * 32;

    v8f acc0 = {}; v8f acc1 = {};
    const int bcol0 = col0 + lrow;
    const int bcol1 = col0 + 16 + lrow;

    for (int k = 0; k < KK; k += 4) {
        v2f a;
        a.x = sA[lrow][k + koff];
        a.y = sA[lrow][k + koff + 1];
        const float* wp = W + (size_t)(k + koff) * DD;
        v2f b0; b0.x = wp[bcol0]; b0.y = wp[DD + bcol0];
        v2f b1; b1.x = wp[bcol1]; b1.y = wp[DD + bcol1];
        acc0 = __builtin_amdgcn_wmma_f32_16x16x4_f32(false, a, false, b0,
                                                     (short)0, acc0, false, false);
        acc1 = __builtin_amdgcn_wmma_f32_16x16x4_f32(false, a, false, b1,
                                                     (short)0, acc1, false, false);
    }

    const int rbase = (lane < 16) ? 0 : 8;
#pragma unroll
    for (int r = 0; r < 8; ++r) {
        int rr = rbase + r;
        zs[rr][col0 + lrow]      = acc0[r] + bias[col0 + lrow];
        zs[rr][col0 + 16 + lrow] = acc1[r] + bias[col0 + 16 + lrow];
    }
    __syncthreads();

    const int lr = t >> 4;
    const int lg = t & 15;
    float s1 = 0.f, s2 = 0.f;
#pragma unroll
    for (int j = 0; j < 16; ++j) {
        float v = zs[lr][lg * 16 + j];
        s1 += v; s2 += v * v;
    }
    red1[lr][lg] = s1; red2[lr][lg] = s2;
    __syncthreads();

    float m1 = 0.f, m2 = 0.f;
#pragma unroll
    for (int j = 0; j < 16; ++j) { m1 += red1[lr][j]; m2 += red2[lr][j]; }
    float mu   = m1 * (1.0f / DD);
    float var  = m2 * (1.0f / DD) - mu * mu;
    float rstd = rsqrtf(var + EPS_LN);

    size_t orow = (size_t)(row0 + lr) * DD;
#pragma unroll
    for (int j = 0; j < 16; ++j) {
        int c = lg * 16 + j;
        float v = (zs[lr][c] - mu) * rstd * gamma[c] + beta[c];
        hout[orow + c] = fmaxf(v, 0.0f);
    }
}

// ===========================================================================
// Output reduction
// ===========================================================================

__global__ __launch_bounds__(256)
void colsum_kernel(const float* __restrict__ m,
                   float* __restrict__ colsum,
                   int rows, int rows_per_block) {
    int c  = threadIdx.x;
    int r0 = blockIdx.x * rows_per_block;
    int r1 = r0 + rows_per_block;
    if (r1 > rows) r1 = rows;
    float acc = 0.f;
    for (int r = r0; r < r1; ++r) acc += m[(size_t)r * DD + c];
    atomic_add_f32(&colsum[c], acc);
}

__global__ __launch_bounds__(256)
void final_kernel(const float* __restrict__ colsum_x,
                  const float* __restrict__ colsum_h,
                  const float* __restrict__ W_skip,   // 256 x 256
                  const float* __restrict__ b_skip,
                  float* __restrict__ out) {
    __shared__ float xm[DD];
    int c = threadIdx.x;
    xm[c] = colsum_x[c] * (1.0f / NN);
    __syncthreads();
    float acc = 0.f;
    for (int k = 0; k < DD; ++k) acc += xm[k] * W_skip[(size_t)k * DD + c];
    out[c] = colsum_h[c] * (1.0f / NN) + acc + b_skip[c];
}

// ===========================================================================
extern "C" void kernel_launch(void* const* d_in, const int* in_sizes, int n_in,
                              void* d_out, int out_size, void* d_ws, size_t ws_size,
                              hipStream_t stream) {
    const float* x        = (const float*)d_in[0];
    const int*   eidx     = (const int*)d_in[1];
    const float* W_sage   = (const float*)d_in[2];   // L x 512 x 256
    const float* b_sage   = (const float*)d_in[3];   // L x 256
    const float* gamma    = (const float*)d_in[4];   // L x 256
    const float* beta     = (const float*)d_in[5];   // L x 256
    const float* W_attn   = (const float*)d_in[6];   // L x 512
    const float* b_attn   = (const float*)d_in[7];   // L
    const float* W_skip   = (const float*)d_in[8];   // 256 x 256
    const float* b_skip   = (const float*)d_in[9];   // 256
    float* out = (float*)d_out;

    const int* src = eidx;        // edge_index[0]
    const int* dst = eidx + EE;   // edge_index[1]

    // ---- workspace layout ----
    float* ws     = (float*)d_ws;
    float* hA     = ws;                         // N*256
    float* hB     = hA + (size_t)NN * DD;       // N*256
    float* agg    = hB + (size_t)NN * DD;       // N*256
    float* s_dst  = agg + (size_t)NN * DD;      // N
    float* s_src  = s_dst + NN;                 // N
    float* a_edge = s_src + NN;                 // E
    float* csx    = a_edge + EE;                // 256
    float* csh    = csx + DD;                   // 256
    int*   deg    = (int*)(csh + DD);           // N
    int*   rowptr = deg + NN;                   // N+1
    int*   csum   = rowptr + NN + 1;            // 256 (>= NCHUNKS)
    int*   pos    = csum + SCAN_CHUNK;          // N
    int*   eid    = pos + NN;                   // E

    const int eblk  = (EE + 255) / 256;
    const int nwblk = (NN + 7) / 8;             // 1 node per wave32
    const int gemmb = NN / 16;                  // 3125 exact

    // ---- CSR build (amortized over the 3 layers) ----
    hipMemsetAsync(deg, 0, NN * sizeof(int), stream);
    count_deg_kernel<<<eblk, 256, 0, stream>>>(dst, deg, EE);
    scan1_kernel<<<NCHUNKS, 256, 0, stream>>>(deg, rowptr, csum, NN);
    scan2_kernel<<<1, 256, 0, stream>>>(csum, NCHUNKS);
    scan3_kernel<<<(NN + 255) / 256, 256, 0, stream>>>(rowptr, csum, NN, EE);
    hipMemcpyAsync(pos, rowptr, NN * sizeof(int), hipMemcpyDeviceToDevice, stream);
    fill_csr_kernel<<<eblk, 256, 0, stream>>>(dst, pos, eid, EE);

    // ---- skip path: mean(x) column sums ----
    hipMemsetAsync(csx, 0, DD * sizeof(float), stream);
    colsum_kernel<<<200, 256, 0, stream>>>(x, csx, NN, 250);

    // ---- layers ----
    const float* h_in = x;
    float* h_out = hA;
    for (int i = 0; i < LL; ++i) {
        attn_scores_kernel<<<nwblk, 256, 0, stream>>>(
            h_in, W_attn + (size_t)i * KK, s_dst, s_src, NN);

        edge_exp_kernel<<<eblk, 256, 0, stream>>>(
            src, dst, s_dst, s_src, b_attn, i, a_edge, EE);

        node_agg_kernel<<<nwblk, 256, 0, stream>>>(
            rowptr, eid, src, a_edge, h_in, agg, NN);

        sage_gemm_ln_kernel<<<gemmb, 256, 0, stream>>>(
            h_in, agg,
            W_sage + (size_t)i * KK * DD,
            b_sage + (size_t)i * DD,
            gamma  + (size_t)i * DD,
            beta   + (size_t)i * DD,
            h_out, NN);

        h_in  = h_out;
        h_out = (h_out == hA) ? hB : hA;
    }

    // ---- output ----
    hipMemsetAsync(csh, 0, DD * sizeof(float), stream);
    colsum_kernel<<<200, 256, 0, stream>>>(h_in, csh, NN, 250);
    final_kernel<<<1, 256, 0, stream>>>(csx, csh, W_skip, b_skip, out);
}